// AGCA_46454366273924
// MI455X (gfx1250) — compile-verified
//
#include <hip/hip_runtime.h>
#include <math.h>

typedef __attribute__((ext_vector_type(2))) float v2f;
typedef __attribute__((ext_vector_type(8))) float v8f;

#define BB   32
#define CIN  256
#define HIDE 64
#define HW   3136   // 56*56

// ---------------------------------------------------------------------------
// Kernel 1: global average pool. One 256-thread block per (b,c) row of 3136
// floats. Coalesced float4 reads (3136 % 4 == 0), LDS tree reduction.
// ---------------------------------------------------------------------------
__global__ __launch_bounds__(256)
void agca_pool_kernel(const float* __restrict__ x, float* __restrict__ y) {
  __shared__ float red[256];
  const int row = blockIdx.x;                       // b*CIN + c
  const float4* xr = (const float4*)(x + (size_t)row * HW);
  float s = 0.f;
  for (int i = threadIdx.x; i < HW / 4; i += 256) {
    float4 v = xr[i];
    s += (v.x + v.y) + (v.z + v.w);
  }
  red[threadIdx.x] = s;
  __syncthreads();
  for (int off = 128; off > 0; off >>= 1) {
    if (threadIdx.x < off) red[threadIdx.x] += red[threadIdx.x + off];
    __syncthreads();
  }
  if (threadIdx.x == 0) y[row] = red[0] * (1.0f / (float)HW);
}

// ---------------------------------------------------------------------------
// Kernel 2: the whole gate pipeline in ONE workgroup (8 waves), GEMMs via
// V_WMMA_F32_16X16X4_F32.
//   Y1 = Y @ W1^T                 M=32 N=64  K=256
//   a  = softmax(w2*Y1, dim=-1)
//   Z  = relu(w3*(Y1*a + Y1@A2))  M=32 N=64  K=64
//   G  = sigmoid(Z @ W4^T)        M=32 N=256 K=64
//
// f32 WMMA wave32 operand layout (ISA 7.12.2):
//   A 16x4 : lane<16 -> {A[m,k0],A[m,k0+1]}, lane>=16 -> {A[m,k0+2],A[m,k0+3]}
//   B 4x16 : lane<16 -> {B[k0,n],B[k0+1,n]}, lane>=16 -> {B[k0+2,n],B[k0+3,n]}
//   C/D    : VGPR v = row v (lanes 0-15) / row v+8 (lanes 16-31), col = lane%16
// ---------------------------------------------------------------------------
__global__ __launch_bounds__(256)
void agca_gate_kernel(const float* __restrict__ y,    // [32,256]
                      const float* __restrict__ W1,   // [64,256]
                      const float* __restrict__ w2p,  // [1]
                      const float* __restrict__ w3p,  // [1]
                      const float* __restrict__ A2,   // [64,64]
                      const float* __restrict__ W4,   // [256,64]
                      float* __restrict__ g)          // [32,256]
{
  __shared__ float sY [BB * CIN];   // 32 KB
  __shared__ float sY1[BB * HIDE];  //  8 KB
  __shared__ float sA [BB * HIDE];  //  8 KB
  __shared__ float sZ [BB * HIDE];  //  8 KB

  const int tid  = threadIdx.x;
  const int wave = tid >> 5;        // 0..7 (wave32)
  const int lane = tid & 31;
  const int half = lane >> 4;       // 0: lanes 0-15, 1: lanes 16-31
  const int l    = lane & 15;

  for (int i = tid; i < BB * CIN; i += 256) sY[i] = y[i];
  __syncthreads();

  // ---- GEMM1: Y1 = Y @ W1^T  (B[k,n] = W1[n,k]) ---------------------------
  {
    const int m0 = (wave >> 2) * 16;
    const int n0 = (wave & 3) * 16;
    v8f acc = {0.f, 0.f, 0.f, 0.f, 0.f, 0.f, 0.f, 0.f};
    for (int k0 = 0; k0 < CIN; k0 += 4) {
      const int ka = k0 + 2 * half;
      v2f a, b;
      a[0] = sY[(m0 + l) * CIN + ka];
      a[1] = sY[(m0 + l) * CIN + ka + 1];
      b[0] = W1[(n0 + l) * CIN + ka];
      b[1] = W1[(n0 + l) * CIN + ka + 1];
      acc = __builtin_amdgcn_wmma_f32_16x16x4_f32(false, a, false, b,
                                                  (short)0, acc, false, false);
    }
#pragma unroll
    for (int v = 0; v < 8; ++v)
      sY1[(m0 + v + 8 * half) * HIDE + (n0 + l)] = acc[v];
  }
  __syncthreads();

  // ---- softmax(w2 * Y1) per row, serial per-thread (tiny: 32 rows x 64) ---
  if (tid < BB) {
    const float w2 = w2p[0];
    float m = -3.4e38f;
    for (int h = 0; h < HIDE; ++h) m = fmaxf(m, w2 * sY1[tid * HIDE + h]);
    float sum = 0.f;
    for (int h = 0; h < HIDE; ++h) {
      float e = __expf(w2 * sY1[tid * HIDE + h] - m);
      sA[tid * HIDE + h] = e;
      sum += e;
    }
    const float inv = 1.0f / sum;
    for (int h = 0; h < HIDE; ++h) sA[tid * HIDE + h] *= inv;
  }
  __syncthreads();

  // ---- GEMM2: Z = relu(w3*(Y1*a + Y1@A2))  (B[k,n] = A2[k,n]) ------------
  {
    const float w3 = w3p[0];
    const int m0 = (wave >> 2) * 16;
    const int n0 = (wave & 3) * 16;
    v8f acc = {0.f, 0.f, 0.f, 0.f, 0.f, 0.f, 0.f, 0.f};
    for (int k0 = 0; k0 < HIDE; k0 += 4) {
      const int ka = k0 + 2 * half;
      v2f a, b;
      a[0] = sY1[(m0 + l) * HIDE + ka];
      a[1] = sY1[(m0 + l) * HIDE + ka + 1];
      b[0] = A2[ka * HIDE + (n0 + l)];
      b[1] = A2[(ka + 1) * HIDE + (n0 + l)];
      acc = __builtin_amdgcn_wmma_f32_16x16x4_f32(false, a, false, b,
                                                  (short)0, acc, false, false);
    }
#pragma unroll
    for (int v = 0; v < 8; ++v) {
      const int r = m0 + v + 8 * half;
      const int c = n0 + l;
      const float y1 = sY1[r * HIDE + c];
      const float zv = w3 * (y1 * sA[r * HIDE + c] + acc[v]);
      sZ[r * HIDE + c] = fmaxf(zv, 0.f);
    }
  }
  __syncthreads();

  // ---- GEMM3: G = sigmoid(Z @ W4^T)  (B[k,n] = W4[n,k]); 32 tiles / 8 waves
  for (int t = wave; t < 32; t += 8) {
    const int m0 = (t >> 4) * 16;
    const int n0 = (t & 15) * 16;
    v8f acc = {0.f, 0.f, 0.f, 0.f, 0.f, 0.f, 0.f, 0.f};
    for (int k0 = 0; k0 < HIDE; k0 += 4) {
      const int ka = k0 + 2 * half;
      v2f a, b;
      a[0] = sZ[(m0 + l) * HIDE + ka];
      a[1] = sZ[(m0 + l) * HIDE + ka + 1];
      b[0] = W4[(n0 + l) * HIDE + ka];
      b[1] = W4[(n0 + l) * HIDE + ka + 1];
      acc = __builtin_amdgcn_wmma_f32_16x16x4_f32(false, a, false, b,
                                                  (short)0, acc, false, false);
    }
#pragma unroll
    for (int v = 0; v < 8; ++v) {
      const int r = m0 + v + 8 * half;
      const int c = n0 + l;
      g[r * CIN + c] = 1.0f / (1.0f + __expf(-acc[v]));
    }
  }
}

// ---------------------------------------------------------------------------
// Kernel 3: out = x * g[b,c]. float4 coalesced; each float4 stays inside one
// (b,c) row since HW % 4 == 0. x should largely hit in the 192MB L2 (warmed
// by the pooling pass).
// ---------------------------------------------------------------------------
__global__ __launch_bounds__(256)
void agca_scale_kernel(const float* __restrict__ x, const float* __restrict__ g,
                       float* __restrict__ out) {
  const size_t i4 = (size_t)blockIdx.x * 256 + threadIdx.x;
  const float gv = g[(i4 * 4) / HW];
  float4 v = ((const float4*)x)[i4];
  v.x *= gv; v.y *= gv; v.z *= gv; v.w *= gv;
  ((float4*)out)[i4] = v;
}

extern "C" void kernel_launch(void* const* d_in, const int* in_sizes, int n_in,
                              void* d_out, int out_size, void* d_ws, size_t ws_size,
                              hipStream_t stream) {
  const float* x  = (const float*)d_in[0];
  const float* W1 = (const float*)d_in[1];
  const float* w2 = (const float*)d_in[2];
  const float* w3 = (const float*)d_in[3];
  const float* A2 = (const float*)d_in[4];
  const float* W4 = (const float*)d_in[5];
  float* out = (float*)d_out;

  float* y = (float*)d_ws;        // [32*256] pooled means
  float* g = y + BB * CIN;        // [32*256] sigmoid gate

  agca_pool_kernel <<<BB * CIN, 256, 0, stream>>>(x, y);
  agca_gate_kernel <<<1,        256, 0, stream>>>(y, W1, w2, w3, A2, W4, g);
  agca_scale_kernel<<<(BB * CIN * HW) / (4 * 256), 256, 0, stream>>>(x, g, out);
}